// EfficientAttention_14396730376599
// MI455X (gfx1250) — compile-verified
//
#include <hip/hip_runtime.h>

// ---------------------------------------------------------------------------
// EfficientAttention for MI455X (gfx1250, wave32, WMMA f32 16x16x4)
// Memory-bound (~440MB HBM traffic, 13 GFLOP): fuse everything, stream inputs
// once, never materialize the [64,HW] attention output in HBM. Double-buffered
// GLOBAL_LOAD_ASYNC_TO_LDS staging (ASYNCcnt pipeline) overlaps global->LDS
// movement with WMMA compute.
// ---------------------------------------------------------------------------

typedef float v2f __attribute__((ext_vector_type(2)));
typedef float v8f __attribute__((ext_vector_type(8)));
typedef int   v4i __attribute__((vector_size(4 * sizeof(int))));

#define NB     8
#define CIN    256
#define HWN    16384
#define KCH    32
#define VCH    64
#define HEADS  8
#define DK     4
#define DV     8
#define COUT   256
#define QSCALE 0.5f   // DK^-0.5

#define TILE_N 128
#define KC     32                 // K staging chunk (double buffered)
#define NCHUNK (CIN / KC)         // 8
#define XS_STRIDE 136             // 128+8: half-wave row offset -> disjoint banks
#define WS_STRIDE 36              // 36m+c distinct mod 64 over 16 lanes; halves differ mod 4
#define KV_STRIDE 132
#define W2S_STRIDE 36
#define XSZ (KC * XS_STRIDE)      // 4352 floats
#define WSZ_KV (96 * WS_STRIDE)   // 3456 floats
#define WSZ_Q  (KCH * WS_STRIDE)  // 1152 floats

// ---- CDNA5 async global->LDS path (guarded; falls back to sync staging) ----
#if __has_builtin(__builtin_amdgcn_global_load_async_to_lds_b128) && \
    __has_builtin(__builtin_amdgcn_s_wait_asynccnt)
#define USE_ASYNC 1
#define ASYNC_COPY16(gp, lp)                                              \
    __builtin_amdgcn_global_load_async_to_lds_b128(                       \
        (__attribute__((address_space(1))) v4i*)(gp),                     \
        (__attribute__((address_space(3))) v4i*)(lp), 0, 0)
#define ASYNC_WAIT(n) __builtin_amdgcn_s_wait_asynccnt(n)
#else
#define USE_ASYNC 0
#define ASYNC_WAIT(n)
#endif

static __device__ __forceinline__ v2f ldA(const float* base, int stride, int m, int kk, int hi) {
    const float* p = base + m * stride + kk;
    v2f a; a.x = p[hi ? 2 : 0]; a.y = p[hi ? 3 : 1]; return a;
}
static __device__ __forceinline__ v2f ldB(const float* base, int stride, int kk, int col, int hi) {
    v2f b2;
    b2.x = base[(kk + (hi ? 2 : 0)) * stride + col];
    b2.y = base[(kk + (hi ? 3 : 1)) * stride + col];
    return b2;
}
#define WMMA_F32(A, B2, C) \
    __builtin_amdgcn_wmma_f32_16x16x4_f32(false, (A), false, (B2), (short)0, (C), false, false)

// --------------------------------------------------------------------------
// K0: zero context accumulators (harness does not re-zero ws per replay)
// --------------------------------------------------------------------------
__global__ __launch_bounds__(256) void k0_zero(float* __restrict__ ctx_g,
                                               float* __restrict__ den_g) {
    int t = threadIdx.x;
    for (int i = t; i < NB * HEADS * DK * DV; i += 256) ctx_g[i] = 0.f;
    for (int i = t; i < NB * HEADS * DK;      i += 256) den_g[i] = 0.f;
}

// --------------------------------------------------------------------------
// K1: k/v projection GEMM [96,256]x[256,128-tile] + exp(k) + context
//     numerator/denominator accumulation. grid = (HW/128, B), 8 waves/block.
//     Double-buffered async staging: 7 async instrs / wave / chunk.
// --------------------------------------------------------------------------
__global__ __launch_bounds__(256)
void k1_proj_kv_ctx(const float* __restrict__ input,
                    const float* __restrict__ kw, const float* __restrict__ kb,
                    const float* __restrict__ vw, const float* __restrict__ vb,
                    float* __restrict__ ctx_g, float* __restrict__ den_g) {
    __shared__ __align__(16) float smem[2 * XSZ + 2 * WSZ_KV]; // 15616 f = 61KB
    // buffer pointers computed arithmetically (no static pointer aggregates:
    // lld cannot relocate addrspacecast-of-LDS in constant initializers)
    float* kvs = smem;  // 96*132 = 12672 <= 15616 (reuse after GEMM)

    const int tid = threadIdx.x;
    const int lane = tid & 31, wave = tid >> 5;
    const int l15 = lane & 15, hi = lane >> 4;
    const int b  = blockIdx.y;
    const int n0 = blockIdx.x * TILE_N;

    const float* xbase = input + (size_t)b * CIN * HWN + n0;

    auto issueX = [&](int buf, int kc) {
        float* dst = smem + buf * XSZ;
#if USE_ASYNC
        // 32 rows x 512B; wave w copies rows 4w..4w+3, each lane 16B of a row
#pragma unroll
        for (int rr = 0; rr < 4; ++rr) {
            int r = wave * 4 + rr;
            ASYNC_COPY16(xbase + (size_t)(kc + r) * HWN + lane * 4,
                         dst + r * XS_STRIDE + lane * 4);
        }
#else
        for (int idx = tid; idx < KC * (TILE_N / 4); idx += 256) {
            int r = idx >> 5, c4 = idx & 31;
            float4 v = *(const float4*)(xbase + (size_t)(kc + r) * HWN + c4 * 4);
            float* d = dst + r * XS_STRIDE + c4 * 4;
            d[0] = v.x; d[1] = v.y; d[2] = v.z; d[3] = v.w;
        }
#endif
    };
    auto issueW = [&](int buf, int kc) {
        float* dst = smem + 2 * XSZ + buf * WSZ_KV;
#if USE_ASYNC
        // 96 rows x 128B; 4 rows per instr; wave w covers rows 12w..12w+11
#pragma unroll
        for (int ii = 0; ii < 3; ++ii) {
            int r = wave * 12 + ii * 4 + (lane >> 3);
            int c = (lane & 7) * 4;
            const float* src = (r < KCH) ? (kw + r * CIN) : (vw + (r - KCH) * CIN);
            ASYNC_COPY16(src + kc + c, dst + r * WS_STRIDE + c);
        }
#else
        for (int idx = tid; idx < 96 * (KC / 4); idx += 256) {
            int r = idx >> 3, c4 = idx & 7;
            const float* src = (r < KCH) ? (kw + r * CIN + kc) : (vw + (r - KCH) * CIN + kc);
            float4 v = *(const float4*)(src + c4 * 4);
            float* d = dst + r * WS_STRIDE + c4 * 4;
            d[0] = v.x; d[1] = v.y; d[2] = v.z; d[3] = v.w;
        }
#endif
    };

    v8f acc[6];
#pragma unroll
    for (int i = 0; i < 6; ++i) acc[i] = (v8f){0, 0, 0, 0, 0, 0, 0, 0};

    issueX(0, 0);
    issueW(0, 0);
    for (int c = 0; c < NCHUNK; ++c) {
        const int cur = c & 1;
        if (c + 1 < NCHUNK) {
            issueX(cur ^ 1, (c + 1) * KC);
            issueW(cur ^ 1, (c + 1) * KC);
            ASYNC_WAIT(7);   // in-order drain: chunk c complete, c+1 in flight
        } else {
            ASYNC_WAIT(0);
        }
        __syncthreads();
        const float* X = smem + cur * XSZ;
        const float* W = smem + 2 * XSZ + cur * WSZ_KV;
        const int nw = wave * 16 + l15;
#pragma unroll
        for (int kk = 0; kk < KC; kk += 4) {
            v2f bf = ldB(X, XS_STRIDE, kk, nw, hi);
#pragma unroll
            for (int mt = 0; mt < 6; ++mt) {
                v2f af = ldA(W, WS_STRIDE, mt * 16 + l15, kk, hi);
                acc[mt] = WMMA_F32(af, bf, acc[mt]);
            }
        }
        __syncthreads();
    }

    // write tile to LDS: k rows stored as exp(k + kb), v rows as v + vb
#pragma unroll
    for (int mt = 0; mt < 6; ++mt) {
#pragma unroll
        for (int j = 0; j < 8; ++j) {
            int row = mt * 16 + j + 8 * hi;
            float val;
            if (row < KCH) val = __expf(acc[mt][j] + kb[row]);
            else           val = acc[mt][j] + vb[row - KCH];
            kvs[row * KV_STRIDE + wave * 16 + l15] = val;
        }
    }
    __syncthreads();

    // thread tid = (h,d,e): ctx numerator over this block's 128 columns
    {
        int e = tid & 7;
        const float* krow = kvs + (tid >> 3) * KV_STRIDE;              // (h*4+d)
        const float* vrow = kvs + (KCH + (tid >> 5) * DV + e) * KV_STRIDE;
        float s = 0.f, sden = 0.f;
#pragma unroll 4
        for (int c = 0; c < TILE_N; ++c) {
            float ek = krow[c];
            s += ek * vrow[c];
            sden += ek;
        }
        atomicAdd(&ctx_g[b * HEADS * DK * DV + tid], s);
        if (e == 0) atomicAdd(&den_g[b * HEADS * DK + (tid >> 3)], sden);
    }
}

// --------------------------------------------------------------------------
// K2: q projection GEMM [32,256]x[256,128-tile] + in-register DK=4 softmax.
//     Double-buffered async staging: 5 async instrs / wave / chunk.
// --------------------------------------------------------------------------
__global__ __launch_bounds__(256)
void k2_proj_q(const float* __restrict__ target,
               const float* __restrict__ qw, const float* __restrict__ qb,
               float* __restrict__ qprime) {
    __shared__ __align__(16) float Ts[2 * XSZ];    // 8704 f
    __shared__ __align__(16) float Wq[2 * WSZ_Q];  // 2304 f

    const int tid = threadIdx.x;
    const int lane = tid & 31, wave = tid >> 5;
    const int l15 = lane & 15, hi = lane >> 4;
    const int b  = blockIdx.y;
    const int n0 = blockIdx.x * TILE_N;

    const float* tbase = target + (size_t)b * CIN * HWN + n0;

    auto issueX = [&](int buf, int kc) {
        float* dst = Ts + buf * XSZ;
#if USE_ASYNC
#pragma unroll
        for (int rr = 0; rr < 4; ++rr) {
            int r = wave * 4 + rr;
            ASYNC_COPY16(tbase + (size_t)(kc + r) * HWN + lane * 4,
                         dst + r * XS_STRIDE + lane * 4);
        }
#else
        for (int idx = tid; idx < KC * (TILE_N / 4); idx += 256) {
            int r = idx >> 5, c4 = idx & 31;
            float4 v = *(const float4*)(tbase + (size_t)(kc + r) * HWN + c4 * 4);
            float* d = dst + r * XS_STRIDE + c4 * 4;
            d[0] = v.x; d[1] = v.y; d[2] = v.z; d[3] = v.w;
        }
#endif
    };
    auto issueW = [&](int buf, int kc) {
        float* dst = Wq + buf * WSZ_Q;
#if USE_ASYNC
        // 32 rows x 128B; 4 rows per instr; wave w covers rows 4w..4w+3
        int r = wave * 4 + (lane >> 3);
        int c = (lane & 7) * 4;
        ASYNC_COPY16(qw + r * CIN + kc + c, dst + r * WS_STRIDE + c);
#else
        for (int idx = tid; idx < KCH * (KC / 4); idx += 256) {
            int r = idx >> 3, c4 = idx & 7;
            float4 v = *(const float4*)(qw + r * CIN + kc + c4 * 4);
            float* d = dst + r * WS_STRIDE + c4 * 4;
            d[0] = v.x; d[1] = v.y; d[2] = v.z; d[3] = v.w;
        }
#endif
    };

    v8f acc[2];
    acc[0] = (v8f){0, 0, 0, 0, 0, 0, 0, 0};
    acc[1] = (v8f){0, 0, 0, 0, 0, 0, 0, 0};

    issueX(0, 0);
    issueW(0, 0);
    for (int c = 0; c < NCHUNK; ++c) {
        const int cur = c & 1;
        if (c + 1 < NCHUNK) {
            issueX(cur ^ 1, (c + 1) * KC);
            issueW(cur ^ 1, (c + 1) * KC);
            ASYNC_WAIT(5);
        } else {
            ASYNC_WAIT(0);
        }
        __syncthreads();
        const float* X = Ts + cur * XSZ;
        const float* W = Wq + cur * WSZ_Q;
        const int nw = wave * 16 + l15;
#pragma unroll
        for (int kk = 0; kk < KC; kk += 4) {
            v2f bf = ldB(X, XS_STRIDE, kk, nw, hi);
#pragma unroll
            for (int mt = 0; mt < 2; ++mt) {
                v2f af = ldA(W, WS_STRIDE, mt * 16 + l15, kk, hi);
                acc[mt] = WMMA_F32(af, bf, acc[mt]);
            }
        }
        __syncthreads();
    }

    // epilogue: rows in this lane = mt*16 + j + 8*hi; elements {0..3} and
    // {4..7} are each one head's complete DK group -> local softmax.
    const int col = n0 + wave * 16 + l15;
#pragma unroll
    for (int mt = 0; mt < 2; ++mt) {
        float vals[8];
#pragma unroll
        for (int j = 0; j < 8; ++j) {
            int row = mt * 16 + j + 8 * hi;
            vals[j] = (acc[mt][j] + qb[row]) * QSCALE;
        }
#pragma unroll
        for (int g = 0; g < 8; g += 4) {
            float m = fmaxf(fmaxf(vals[g], vals[g + 1]), fmaxf(vals[g + 2], vals[g + 3]));
            float e0 = __expf(vals[g] - m),     e1 = __expf(vals[g + 1] - m);
            float e2 = __expf(vals[g + 2] - m), e3 = __expf(vals[g + 3] - m);
            float inv = 1.f / (e0 + e1 + e2 + e3);
            vals[g] = e0 * inv; vals[g + 1] = e1 * inv;
            vals[g + 2] = e2 * inv; vals[g + 3] = e3 * inv;
        }
#pragma unroll
        for (int j = 0; j < 8; ++j) {
            int row = mt * 16 + j + 8 * hi;
            qprime[((size_t)b * KCH + row) * HWN + col] = vals[j];
        }
    }
}

// --------------------------------------------------------------------------
// K3: normalize context and fold into output projection:
//     W2[b][o][h*4+d] = sum_e rw[o][h*8+e] * (ctx[h,d,e]/den[h,d])
// --------------------------------------------------------------------------
__global__ __launch_bounds__(256)
void k3_fold(const float* __restrict__ ctx_g, const float* __restrict__ den_g,
             const float* __restrict__ rw, float* __restrict__ w2) {
    __shared__ float C[HEADS * DK * DV];
    const int b = blockIdx.x, tid = threadIdx.x;
    if (tid < HEADS * DK * DV) {
        int hd = tid >> 3;  // h*4+d
        C[tid] = ctx_g[b * HEADS * DK * DV + tid] / den_g[b * HEADS * DK + hd];
    }
    __syncthreads();
    const int o = tid;  // output channel 0..255
    float rwr[VCH];
#pragma unroll 8
    for (int c = 0; c < VCH; ++c) rwr[c] = rw[o * VCH + c];
#pragma unroll
    for (int h = 0; h < HEADS; ++h) {
#pragma unroll
        for (int d = 0; d < DK; ++d) {
            float s = 0.f;
#pragma unroll
            for (int e = 0; e < DV; ++e) s += rwr[h * DV + e] * C[(h * DK + d) * DV + e];
            w2[((size_t)b * COUT + o) * KCH + h * DK + d] = s;
        }
    }
}

// --------------------------------------------------------------------------
// K4: result[b] = W2[b] ([256,32]) x q' ([32,128-tile]) + rb -> 134MB stream.
// --------------------------------------------------------------------------
__global__ __launch_bounds__(256)
void k4_out(const float* __restrict__ qprime, const float* __restrict__ w2,
            const float* __restrict__ rb, float* __restrict__ out) {
    __shared__ __align__(16) float W2s[COUT * W2S_STRIDE]; // 9216 f
    __shared__ __align__(16) float Qs[KCH * XS_STRIDE];    // 4352 f

    const int tid = threadIdx.x;
    const int lane = tid & 31, wave = tid >> 5;
    const int l15 = lane & 15, hi = lane >> 4;
    const int b  = blockIdx.y;
    const int n0 = blockIdx.x * TILE_N;

#if USE_ASYNC
    // W2: 256 rows x 128B; 4 rows/instr; wave w covers rows 32w..32w+31
#pragma unroll
    for (int ii = 0; ii < 8; ++ii) {
        int r = (wave * 8 + ii) * 4 + (lane >> 3);
        int c = (lane & 7) * 4;
        ASYNC_COPY16(w2 + ((size_t)b * COUT + r) * KCH + c, W2s + r * W2S_STRIDE + c);
    }
    // q' tile: 32 rows x 512B; wave w copies rows 4w..4w+3
#pragma unroll
    for (int rr = 0; rr < 4; ++rr) {
        int r = wave * 4 + rr;
        ASYNC_COPY16(qprime + ((size_t)b * KCH + r) * HWN + n0 + lane * 4,
                     Qs + r * XS_STRIDE + lane * 4);
    }
    ASYNC_WAIT(0);
#else
    for (int idx = tid; idx < COUT * (KCH / 4); idx += 256) {
        int r = idx >> 3, c4 = idx & 7;
        float4 v = *(const float4*)(w2 + ((size_t)b * COUT + r) * KCH + c4 * 4);
        float* d = W2s + r * W2S_STRIDE + c4 * 4;
        d[0] = v.x; d[1] = v.y; d[2] = v.z; d[3] = v.w;
    }
    for (int idx = tid; idx < KCH * (TILE_N / 4); idx += 256) {
        int r = idx >> 5, c4 = idx & 31;
        float4 v = *(const float4*)(qprime + ((size_t)b * KCH + r) * HWN + n0 + c4 * 4);
        float* d = Qs + r * XS_STRIDE + c4 * 4;
        d[0] = v.x; d[1] = v.y; d[2] = v.z; d[3] = v.w;
    }
#endif
    __syncthreads();

    v8f acc[16];
#pragma unroll
    for (int i = 0; i < 16; ++i) acc[i] = (v8f){0, 0, 0, 0, 0, 0, 0, 0};

    const int nw = wave * 16 + l15;
#pragma unroll
    for (int kk = 0; kk < KCH; kk += 4) {
        v2f bf = ldB(Qs, XS_STRIDE, kk, nw, hi);
#pragma unroll
        for (int mt = 0; mt < 16; ++mt) {
            v2f af = ldA(W2s, W2S_STRIDE, mt * 16 + l15, kk, hi);
            acc[mt] = WMMA_F32(af, bf, acc[mt]);
        }
    }

    const int col = n0 + wave * 16 + l15;
#pragma unroll
    for (int mt = 0; mt < 16; ++mt) {
#pragma unroll
        for (int j = 0; j < 8; ++j) {
            int row = mt * 16 + j + 8 * hi;
            out[((size_t)b * COUT + row) * HWN + col] = acc[mt][j] + rb[row];
        }
    }
}

// --------------------------------------------------------------------------
extern "C" void kernel_launch(void* const* d_in, const int* in_sizes, int n_in,
                              void* d_out, int out_size, void* d_ws, size_t ws_size,
                              hipStream_t stream) {
    const float* target = (const float*)d_in[0];
    const float* input  = (const float*)d_in[1];
    const float* kw = (const float*)d_in[2];
    const float* kb = (const float*)d_in[3];
    const float* qw = (const float*)d_in[4];
    const float* qb = (const float*)d_in[5];
    const float* vw = (const float*)d_in[6];
    const float* vb = (const float*)d_in[7];
    const float* rw = (const float*)d_in[8];
    const float* rb = (const float*)d_in[9];
    float* out = (float*)d_out;

    // workspace layout (floats)
    float* ws     = (float*)d_ws;
    float* qprime = ws;                                  // B*32*HW = 4,194,304
    float* ctx_g  = qprime + (size_t)NB * KCH * HWN;     // 2048
    float* den_g  = ctx_g + NB * HEADS * DK * DV;        // 256
    float* w2     = den_g + NB * HEADS * DK;             // 65,536

    dim3 blk(256);
    dim3 gtile(HWN / TILE_N, NB);

    k0_zero<<<1, blk, 0, stream>>>(ctx_g, den_g);
    k1_proj_kv_ctx<<<gtile, blk, 0, stream>>>(input, kw, kb, vw, vb, ctx_g, den_g);
    k2_proj_q<<<gtile, blk, 0, stream>>>(target, qw, qb, qprime);
    k3_fold<<<NB, blk, 0, stream>>>(ctx_g, den_g, rw, w2);
    k4_out<<<gtile, blk, 0, stream>>>(qprime, w2, rb, out);
}